// SemanticLayer_25048249270820
// MI455X (gfx1250) — compile-verified
//
#include <hip/hip_runtime.h>
#include <hip/hip_bf16.h>

// GAT-style semantic layer for MI455X (gfx1250, wave32).
//   h = x @ W_lin^T                      -> fp32 WMMA GEMM
//   a_src/a_dst per-node attention dots  -> wave32 shuffle reduction
//   edge scatter (dominant: ~1.5GB L2 atomic traffic)
//   out_i = feats_i @ layers_W[i]^T      -> fp32 WMMA GEMM
// Workspace: N*256 (h) + N*4 + N*4 (att) + 4*N*256 (feats) floats ~ 258MB.

typedef float v2f __attribute__((ext_vector_type(2)));
typedef float v8f __attribute__((ext_vector_type(8)));

#define IN_F   256
#define NHEADS 4
#define DH     64
#define LDSW   68   // 64 + 4 pad floats -> 16 row-strided lanes hit distinct LDS banks

// ---------------------------------------------------------------- zero fill
__global__ void zero_kernel(float4* __restrict__ p, size_t n4) {
  size_t i = (size_t)blockIdx.x * blockDim.x + threadIdx.x;
  if (i < n4) p[i] = make_float4(0.f, 0.f, 0.f, 0.f);
}

// ---------------------------------------------------------------- fp32 WMMA GEMM
// OUT[row, z*out_col_per_z + blockIdx.y*64 + c] = sum_k X_z[row,k] * W_z[blockIdx.y*64 + c, k]
// X row-major [nrows, 256], W row-major [outw, 256]. Block: 128 thr = 4 waves,
// each wave owns a 16-row slab; block tile = 64 rows x 64 cols, K staged via LDS.
__global__ __launch_bounds__(128)
void wmma_matmul_kernel(const float* __restrict__ Xbase, size_t x_z_stride,
                        const float* __restrict__ Wbase, size_t w_z_stride,
                        float* __restrict__ OUT, int out_col_per_z, int nrows) {
  __shared__ float sX[64 * LDSW];
  __shared__ float sW[64 * LDSW];

  const float* X = Xbase + (size_t)blockIdx.z * x_z_stride;
  const float* W = Wbase + (size_t)blockIdx.z * w_z_stride;
  const int colTotal = blockIdx.z * out_col_per_z + blockIdx.y * 64;

  const int tid  = threadIdx.x;
  const int wave = tid >> 5;
  const int lane = tid & 31;
  const int hi   = lane >> 4;   // lane half selects K={0,1} vs K={2,3} of the 16x4 A frag
  const int mrow = lane & 15;
  const int rowBase = blockIdx.x * 64;

  v8f acc[4] = {};  // 4 col-tiles of 16x16 f32 accumulators (8 VGPRs each)

  for (int kt = 0; kt < IN_F; kt += 64) {
    // Cooperative coalesced loads: 64x64 X slab + 64x64 W tile, float4 granularity.
    #pragma unroll
    for (int j = 0; j < 8; ++j) {
      int idx = tid + j * 128;            // 0..1023 float4 slots
      int row = idx >> 4;                 // 16 float4 per row
      int q   = idx & 15;
      int grow = rowBase + row;
      float4 vx = (grow < nrows)
          ? *reinterpret_cast<const float4*>(X + (size_t)grow * IN_F + kt + q * 4)
          : make_float4(0.f, 0.f, 0.f, 0.f);
      *reinterpret_cast<float4*>(&sX[row * LDSW + q * 4]) = vx;
      int wrow = blockIdx.y * 64 + row;   // weight row == output column
      float4 vw = *reinterpret_cast<const float4*>(W + (size_t)wrow * IN_F + kt + q * 4);
      *reinterpret_cast<float4*>(&sW[row * LDSW + q * 4]) = vw;
    }
    __syncthreads();

    // 16 K-steps of V_WMMA_F32_16X16X4_F32, A reused across 4 B col-tiles.
    const float* xrow = &sX[(wave * 16 + mrow) * LDSW + 2 * hi];
    for (int kk = 0; kk < 64; kk += 4) {
      v2f a = *reinterpret_cast<const v2f*>(xrow + kk);   // A[mrow][kk+2*hi .. +1]
      #pragma unroll
      for (int ct = 0; ct < 4; ++ct) {
        v2f b = *reinterpret_cast<const v2f*>(&sW[(ct * 16 + mrow) * LDSW + kk + 2 * hi]);
        acc[ct] = __builtin_amdgcn_wmma_f32_16x16x4_f32(
            false, a, false, b, (short)0, acc[ct], false, false);
      }
    }
    __syncthreads();
  }

  // C/D layout: VGPR r holds row (r + 8*hi), col = mrow. Lanes 0-15 write 64B runs.
  #pragma unroll
  for (int ct = 0; ct < 4; ++ct) {
    #pragma unroll
    for (int r = 0; r < 8; ++r) {
      int row = rowBase + wave * 16 + r + 8 * hi;
      if (row < nrows)
        OUT[(size_t)row * 256 + colTotal + ct * 16 + mrow] = acc[ct][r];
    }
  }
}

// ---------------------------------------------------------------- per-node attention dots
// One wave32 per node: a_src[n,i] = h[n]·atts_W[i,0:256], a_dst[n,i] = h[n]·atts_W[i,256:512]
__global__ __launch_bounds__(256)
void att_dot_kernel(const float* __restrict__ h, const float* __restrict__ attsW,
                    float* __restrict__ a_src, float* __restrict__ a_dst, int n) {
  const int lane  = threadIdx.x & 31;
  const int gwave = (int)((blockIdx.x * (size_t)blockDim.x + threadIdx.x) >> 5);
  if (gwave >= n) return;  // uniform per wave

  float ws[NHEADS][8], wd[NHEADS][8];
  #pragma unroll
  for (int i = 0; i < NHEADS; ++i)
    #pragma unroll
    for (int j = 0; j < 8; ++j) {
      ws[i][j] = attsW[i * 512 + lane * 8 + j];
      wd[i][j] = attsW[i * 512 + 256 + lane * 8 + j];
    }

  const float* hp = h + (size_t)gwave * 256 + lane * 8;
  float4 h0 = *reinterpret_cast<const float4*>(hp);
  float4 h1 = *reinterpret_cast<const float4*>(hp + 4);
  float hv[8] = {h0.x, h0.y, h0.z, h0.w, h1.x, h1.y, h1.z, h1.w};

  float r[8];
  #pragma unroll
  for (int i = 0; i < NHEADS; ++i) {
    float s = 0.f, d = 0.f;
    #pragma unroll
    for (int j = 0; j < 8; ++j) { s += hv[j] * ws[i][j]; d += hv[j] * wd[i][j]; }
    r[i] = s; r[4 + i] = d;
  }
  #pragma unroll
  for (int off = 16; off > 0; off >>= 1)
    #pragma unroll
    for (int k = 0; k < 8; ++k)
      r[k] += __shfl_xor(r[k], off, 32);

  if (lane == 0) {
    a_src[(size_t)gwave * 4 + 0] = r[0]; a_src[(size_t)gwave * 4 + 1] = r[1];
    a_src[(size_t)gwave * 4 + 2] = r[2]; a_src[(size_t)gwave * 4 + 3] = r[3];
    a_dst[(size_t)gwave * 4 + 0] = r[4]; a_dst[(size_t)gwave * 4 + 1] = r[5];
    a_dst[(size_t)gwave * 4 + 2] = r[6]; a_dst[(size_t)gwave * 4 + 3] = r[7];
  }
}

// ---------------------------------------------------------------- edge scatter (bottleneck)
// One 256-thread block per edge: coef[i]=sigmoid(a_src[s,i]+a_dst[d,i]);
// feats[i][s][:] += coef[i]*x[d][:] via global_atomic_add_f32 (L2-resident RMW).
__global__ __launch_bounds__(256)
void edge_scatter_kernel(const int* __restrict__ src, const int* __restrict__ dst,
                         const float* __restrict__ a_src, const float* __restrict__ a_dst,
                         const float* __restrict__ x, float* __restrict__ feats,
                         int n_nodes) {
  __shared__ float coef[NHEADS];
  const int e   = blockIdx.x;
  const int tid = threadIdx.x;
  const int s = src[e];
  const int d = dst[e];
  if (tid < NHEADS) {
    float z = a_src[(size_t)s * 4 + tid] + a_dst[(size_t)d * 4 + tid];
    coef[tid] = 1.f / (1.f + __expf(-z));
  }
  __syncthreads();
  float v = x[(size_t)d * 256 + tid];
  const size_t hstride = (size_t)n_nodes * 256;
  size_t base = (size_t)s * 256 + tid;
  #pragma unroll
  for (int i = 0; i < NHEADS; ++i)
    unsafeAtomicAdd(&feats[base + (size_t)i * hstride], coef[i] * v);
}

// ---------------------------------------------------------------- launch
extern "C" void kernel_launch(void* const* d_in, const int* in_sizes, int n_in,
                              void* d_out, int out_size, void* d_ws, size_t ws_size,
                              hipStream_t stream) {
  const float* x         = (const float*)d_in[0];
  const int*   edge_idx  = (const int*)d_in[1];
  const float* W_lin     = (const float*)d_in[2];
  const float* atts_W    = (const float*)d_in[3];
  const float* layers_W  = (const float*)d_in[4];
  float* out = (float*)d_out;

  const int N = in_sizes[0] / IN_F;
  const int E = in_sizes[1] / 2;
  const int* src = edge_idx;
  const int* dst = edge_idx + E;

  float* ws    = (float*)d_ws;
  float* h     = ws;
  float* asrc  = h + (size_t)N * 256;
  float* adst  = asrc + (size_t)N * NHEADS;
  float* feats = adst + (size_t)N * NHEADS;

  // 1) zero feats accumulator (harness does not re-zero between replays)
  size_t n4 = (size_t)NHEADS * N * 256 / 4;
  zero_kernel<<<(unsigned)((n4 + 255) / 256), 256, 0, stream>>>((float4*)feats, n4);

  // 2) h = x @ W_lin^T   (grid: row tiles x 4 col tiles)
  dim3 gA((N + 63) / 64, IN_F / 64, 1);
  wmma_matmul_kernel<<<gA, 128, 0, stream>>>(x, 0, W_lin, 0, h, 0, N);

  // 3) per-node attention logits (one wave per node)
  att_dot_kernel<<<(N * 32 + 255) / 256, 256, 0, stream>>>(h, atts_W, asrc, adst, N);

  // 4) edge scatter-accumulate
  edge_scatter_kernel<<<E, 256, 0, stream>>>(src, dst, asrc, adst, x, feats, N);

  // 5) out[:, i*64:(i+1)*64] = feats_i @ layers_W[i]^T   (grid.z = heads)
  dim3 gD((N + 63) / 64, 1, NHEADS);
  wmma_matmul_kernel<<<gD, 128, 0, stream>>>(feats, (size_t)N * 256,
                                             layers_W, (size_t)DH * IN_F,
                                             out, DH, N);
}